// KabschRMSDLoss_21852793602126
// MI455X (gfx1250) — compile-verified
//
#include <hip/hip_runtime.h>
#include <hip/hip_bf16.h>

// ---------------------------------------------------------------------------
// Kabsch RMSD loss, MI455X (gfx1250, wave32, WMMA + async-LDS staging).
//
// preds:  [B=256, K=5, N=4096, 3] f32
// target: [B=256, N=4096, 3] f32
// out:    scalar f32
//
// Phase 1 streams all 75.5MB once (HBM-bound, ~3.2us floor @ 23.3TB/s) and
// accumulates, per b, an augmented 16x16 WMMA product:
//   A (16 x Npts): rows 0..14 = P_{k,i} streams (k=0..4, i=x,y,z), row 15 = 1
//   B (Npts x 16): cols 0..2  = Q_{x,y,z}, col 3 = 1, cols 4..15 = 0
//   D[3k+i][j] = sum P_{k,i} Q_j   (uncentered covariance, all 5 k at once)
//   D[3k+i][3] = sum P_{k,i}       (Sp)
//   D[15][j]   = sum Q_j           (Sq)
//   D[15][3]   = point count
// Spp_k and Sqq accumulate in VALU alongside. Phase 2 does the 3x3 SVD math.
// Global->LDS staging uses GLOBAL_LOAD_ASYNC_TO_LDS_B128 when available.
// ---------------------------------------------------------------------------

typedef float v2f __attribute__((ext_vector_type(2)));
typedef float v8f __attribute__((ext_vector_type(8)));
typedef int   v4i __attribute__((ext_vector_type(4)));

#define AS_GLOBAL __attribute__((address_space(1)))
#define AS_LOCAL  __attribute__((address_space(3)))

#if defined(__has_builtin)
#if __has_builtin(__builtin_amdgcn_global_load_async_to_lds_b128)
#define USE_ASYNC_LDS 1
#endif
#endif
#ifndef USE_ASYNC_LDS
#define USE_ASYNC_LDS 0
#endif

#define BDIM   256
#define NB     256          // batches
#define NK     5            // preds per batch
#define NPTS   4096         // points
#define CHUNK  512          // points per block
#define NCHUNK (NPTS / CHUNK)          // 8
#define PFLOATS 80                     // partial floats per block
#define NBLK1  (NB * NCHUNK)           // 2048 phase-1 blocks
#define RMSD_OFF ((size_t)NBLK1 * PFLOATS)

__global__ void __launch_bounds__(BDIM)
kabsch_phase1(const float* __restrict__ gP, const float* __restrict__ gQ,
              float* __restrict__ part) {
    const int blk  = blockIdx.x;
    const int b    = blk >> 3;          // batch
    const int c    = blk & 7;           // chunk
    const int tid  = threadIdx.x;
    const int lane = tid & 31;
    const int w    = tid >> 5;          // wave id, 0..7

    __shared__ __align__(16) float sP[NK * CHUNK * 3];   // 7680 f (30 KB)
    __shared__ __align__(16) float sQ[CHUNK * 3];        // 1536 f ( 6 KB)
    __shared__ float sD[8 * 256];                        // wave accumulators
    __shared__ float sSpp[8 * 32];
    __shared__ float sSqq[8 * 32];

    // ---- stage chunk into LDS: B128 packets (slice bases are 16B-aligned) ----
    const int sliceElems = CHUNK * 3;       // 1536 floats
    const int slicePk    = sliceElems / 4;  // 384 float4 packets
#if USE_ASYNC_LDS
    for (int k = 0; k < NK; ++k) {
        AS_GLOBAL char* src = (AS_GLOBAL char*)
            (gP + ((size_t)(b * NK + k) * NPTS + c * CHUNK) * 3);
        AS_LOCAL char* dst = (AS_LOCAL char*)&sP[k * sliceElems];
        for (int idx = tid; idx < slicePk; idx += BDIM)
            __builtin_amdgcn_global_load_async_to_lds_b128(
                (AS_GLOBAL v4i*)(src + (size_t)idx * 16),
                (AS_LOCAL  v4i*)(dst + (size_t)idx * 16), 0, 0);
    }
    {
        AS_GLOBAL char* src = (AS_GLOBAL char*)
            (gQ + ((size_t)b * NPTS + c * CHUNK) * 3);
        AS_LOCAL char* dst = (AS_LOCAL char*)&sQ[0];
        for (int idx = tid; idx < slicePk; idx += BDIM)
            __builtin_amdgcn_global_load_async_to_lds_b128(
                (AS_GLOBAL v4i*)(src + (size_t)idx * 16),
                (AS_LOCAL  v4i*)(dst + (size_t)idx * 16), 0, 0);
    }
#if __has_builtin(__builtin_amdgcn_s_wait_asynccnt)
    __builtin_amdgcn_s_wait_asynccnt(0);
#else
    asm volatile("s_wait_asynccnt 0x0" ::: "memory");
#endif
#else
    for (int k = 0; k < NK; ++k) {
        const float4* src = (const float4*)
            (gP + ((size_t)(b * NK + k) * NPTS + c * CHUNK) * 3);
        float4* dst = (float4*)&sP[k * sliceElems];
        for (int idx = tid; idx < slicePk; idx += BDIM) dst[idx] = src[idx];
    }
    {
        const float4* src = (const float4*)(gQ + ((size_t)b * NPTS + c * CHUNK) * 3);
        float4* dst = (float4*)&sQ[0];
        for (int idx = tid; idx < slicePk; idx += BDIM) dst[idx] = src[idx];
    }
#endif
    __syncthreads();

    // ---- per-lane WMMA fragment roles ----
    // A frag (16x4 f32): lane L<16 holds A[m][0],A[m][1]; lane L+16 holds
    //   A[m][2],A[m][3]  (m = L&15).  B frag (4x16) mirrors with K split
    //   across lane halves; both halves of a lane consume points q0,q0+1.
    const int half = lane >> 4;            // 0/1 -> K pair
    const int mn   = lane & 15;            // A row / B col
    const bool isOnesRow = (mn == 15);
    const int  kc = (mn / 3 < NK) ? (mn / 3) : (NK - 1);  // clamped for m=15
    const int  ic = (mn - kc * 3 >= 3) ? 0 : (mn - kc * 3);
    const int  nc = (mn < 3) ? mn : 0;     // clamped Q coord for B load
    const float bConst = (mn == 3) ? 1.0f : 0.0f;
    const bool bIsQ = (mn < 3);

    const int wbase = w * 64;              // this wave's 64-point slice
    v8f acc = {0.f, 0.f, 0.f, 0.f, 0.f, 0.f, 0.f, 0.f};
    float spp = 0.f, sqq = 0.f;            // raw; masked once after the loop

    for (int it = 0; it < 16; ++it) {
        const int p0 = wbase + it * 4 + half * 2;
        const int p1 = p0 + 1;

        const float pv0 = sP[kc * sliceElems + p0 * 3 + ic];
        const float pv1 = sP[kc * sliceElems + p1 * 3 + ic];
        const float qv0 = sQ[p0 * 3 + nc];
        const float qv1 = sQ[p1 * 3 + nc];

        const float ax = isOnesRow ? 1.0f : pv0;
        const float ay = isOnesRow ? 1.0f : pv1;
        const float bx = bIsQ ? qv0 : bConst;
        const float by = bIsQ ? qv1 : bConst;

        // raw squares; ones-row / ones-col contributions masked after loop
        spp = fmaf(ax, ax, fmaf(ay, ay, spp));
        sqq = fmaf(bx, bx, fmaf(by, by, sqq));

        v2f a = {ax, ay};
        v2f bb = {bx, by};
        // D(16x16,f32) += A(16x4,f32) x B(4x16,f32)
        acc = __builtin_amdgcn_wmma_f32_16x16x4_f32(
                  false, a, false, bb, (short)0, acc, false, false);
    }
    if (isOnesRow) spp = 0.f;   // row 15 was all 1s
    if (!bIsQ)     sqq = 0.f;   // col 3 was all 1s, cols 4..15 zero

    // ---- spill wave accumulators to LDS in canonical [row][col] layout ----
    // C/D layout: vgpr r -> row r (lanes 0-15) / row r+8 (lanes 16-31).
    #pragma unroll
    for (int r = 0; r < 8; ++r) {
        const int row = r + 8 * half;
        sD[w * 256 + row * 16 + mn] = acc[r];
    }
    sSpp[w * 32 + lane] = spp;
    sSqq[w * 32 + lane] = sqq;
    __syncthreads();

    // ---- block reduction across 8 waves -> 80-float partial ----
    float* outp = part + (size_t)blk * PFLOATS;
    if (tid < 64) {                       // D rows 0..15, cols 0..3
        const int row = tid >> 2, col = tid & 3;
        float s = 0.f;
        #pragma unroll
        for (int ww = 0; ww < 8; ++ww) s += sD[ww * 256 + row * 16 + col];
        outp[tid] = s;
    } else if (tid < 64 + NK) {           // Spp_k
        const int k = tid - 64;
        float s = 0.f;
        for (int ww = 0; ww < 8; ++ww)
            for (int i = 0; i < 3; ++i)
                s += sSpp[ww * 32 + 3 * k + i] + sSpp[ww * 32 + 16 + 3 * k + i];
        outp[tid] = s;
    } else if (tid == 69) {               // Sqq
        float s = 0.f;
        for (int ww = 0; ww < 8; ++ww)
            for (int i = 0; i < 3; ++i)
                s += sSqq[ww * 32 + i] + sSqq[ww * 32 + 16 + i];
        outp[69] = s;
    }
}

__device__ inline float det3f(const float M[3][3]) {
    return M[0][0] * (M[1][1] * M[2][2] - M[1][2] * M[2][1])
         - M[0][1] * (M[1][0] * M[2][2] - M[1][2] * M[2][0])
         + M[0][2] * (M[1][0] * M[2][1] - M[1][1] * M[2][0]);
}

__global__ void __launch_bounds__(BDIM)
kabsch_phase2(const float* __restrict__ part, float* __restrict__ rmsd) {
    const int pid = blockIdx.x * blockDim.x + threadIdx.x;
    if (pid >= NB * NK) return;
    const int b = pid / NK, k = pid % NK;

    float Spq[3][3] = {{0}}, Sp[3] = {0}, Sq[3] = {0};
    float Spp = 0.f, Sqq = 0.f, Ncnt = 0.f;
    for (int c = 0; c < NCHUNK; ++c) {
        const float* pb = part + (size_t)(b * NCHUNK + c) * PFLOATS;
        for (int i = 0; i < 3; ++i) {
            for (int j = 0; j < 3; ++j) Spq[i][j] += pb[(3 * k + i) * 4 + j];
            Sp[i] += pb[(3 * k + i) * 4 + 3];
            Sq[i] += pb[15 * 4 + i];
        }
        Ncnt += pb[63];
        Spp  += pb[64 + k];
        Sqq  += pb[69];
    }

    const float N = Ncnt, invN = 1.f / N;
    float Pm[3], Qm[3];
    for (int i = 0; i < 3; ++i) { Pm[i] = Sp[i] * invN; Qm[i] = Sq[i] * invN; }
    float H[3][3];
    for (int i = 0; i < 3; ++i)
        for (int j = 0; j < 3; ++j)
            H[i][j] = Spq[i][j] - N * Pm[i] * Qm[j];
    const float trP = Spp - N * (Pm[0]*Pm[0] + Pm[1]*Pm[1] + Pm[2]*Pm[2]);
    const float trQ = Sqq - N * (Qm[0]*Qm[0] + Qm[1]*Qm[1] + Qm[2]*Qm[2]);

    // ---- 3x3 SVD via one-sided Jacobi: H = U diag(s) V^T ----
    float A[3][3], V[3][3];
    for (int i = 0; i < 3; ++i)
        for (int j = 0; j < 3; ++j) { A[i][j] = H[i][j]; V[i][j] = (i == j); }

    for (int sweep = 0; sweep < 10; ++sweep) {
        for (int p = 0; p < 2; ++p) {
            for (int q = p + 1; q < 3; ++q) {
                float app = 0.f, aqq = 0.f, apq = 0.f;
                for (int i = 0; i < 3; ++i) {
                    app += A[i][p] * A[i][p];
                    aqq += A[i][q] * A[i][q];
                    apq += A[i][p] * A[i][q];
                }
                if (fabsf(apq) > 1e-20f) {
                    const float zeta = (aqq - app) / (2.f * apq);
                    const float t = copysignf(1.f, zeta) /
                                    (fabsf(zeta) + sqrtf(1.f + zeta * zeta));
                    const float cth = 1.f / sqrtf(1.f + t * t);
                    const float sth = cth * t;
                    for (int i = 0; i < 3; ++i) {
                        const float ap = A[i][p], aq = A[i][q];
                        A[i][p] = cth * ap - sth * aq;
                        A[i][q] = sth * ap + cth * aq;
                        const float vp = V[i][p], vq = V[i][q];
                        V[i][p] = cth * vp - sth * vq;
                        V[i][q] = sth * vp + cth * vq;
                    }
                }
            }
        }
    }

    float s[3];
    for (int j = 0; j < 3; ++j)
        s[j] = sqrtf(A[0][j]*A[0][j] + A[1][j]*A[1][j] + A[2][j]*A[2][j]);
    // sort singular values descending, permuting columns of A and V
    for (int a2 = 0; a2 < 2; ++a2)
        for (int b2 = a2 + 1; b2 < 3; ++b2)
            if (s[b2] > s[a2]) {
                float ts = s[a2]; s[a2] = s[b2]; s[b2] = ts;
                for (int i = 0; i < 3; ++i) {
                    float t1 = A[i][a2]; A[i][a2] = A[i][b2]; A[i][b2] = t1;
                    float t2 = V[i][a2]; V[i][a2] = V[i][b2]; V[i][b2] = t2;
                }
            }
    float U[3][3];
    for (int j = 0; j < 3; ++j) {
        const float inv = 1.f / fmaxf(s[j], 1e-30f);
        for (int i = 0; i < 3; ++i) U[i][j] = A[i][j] * inv;
    }

    // d = det(Vh U^T) = det(V) * det(U);  e = diag(1,1,d)
    const float d = det3f(U) * det3f(V);
    const float e[3] = {1.f, 1.f, d};

    // R = Vh * diag(e) * U^T ;  R[i][j] = sum_a V[a][i] e[a] U[j][a]
    // mse*3N = trP + trQ - 2 tr(R H)
    float trRH = 0.f;
    for (int i = 0; i < 3; ++i)
        for (int j = 0; j < 3; ++j) {
            float Rij = 0.f;
            for (int a2 = 0; a2 < 3; ++a2)
                Rij += V[a2][i] * e[a2] * U[j][a2];
            trRH += Rij * H[j][i];
        }

    const float mse = (trP + trQ - 2.f * trRH) / (3.f * N);
    rmsd[pid] = sqrtf(fmaxf(mse + 1e-8f, 0.f));
}

__global__ void __launch_bounds__(BDIM)
kabsch_phase3(const float* __restrict__ rmsd, float* __restrict__ out) {
    __shared__ float red[BDIM];
    const int b = threadIdx.x;                 // BDIM == NB == 256
    float m = rmsd[b * NK];
    for (int k = 1; k < NK; ++k) m = fminf(m, rmsd[b * NK + k]);
    red[b] = m;
    __syncthreads();
    for (int off = BDIM / 2; off > 0; off >>= 1) {
        if (b < off) red[b] += red[b + off];
        __syncthreads();
    }
    if (b == 0) out[0] = red[0] / (float)NB;
}

extern "C" void kernel_launch(void* const* d_in, const int* in_sizes, int n_in,
                              void* d_out, int out_size, void* d_ws, size_t ws_size,
                              hipStream_t stream) {
    const float* preds  = (const float*)d_in[0];
    const float* target = (const float*)d_in[1];
    float* out  = (float*)d_out;
    float* ws   = (float*)d_ws;
    float* part = ws;                       // NBLK1 * PFLOATS floats
    float* rms  = ws + RMSD_OFF;            // NB * NK floats

    kabsch_phase1<<<dim3(NBLK1), dim3(BDIM), 0, stream>>>(preds, target, part);
    kabsch_phase2<<<dim3((NB * NK + BDIM - 1) / BDIM), dim3(BDIM), 0, stream>>>(part, rms);
    kabsch_phase3<<<dim3(1), dim3(BDIM), 0, stream>>>(rms, out);
}